// LatentSpectralKAFM_88776974009043
// MI455X (gfx1250) — compile-verified
//
#include <hip/hip_runtime.h>
#include <hip/hip_bf16.h>

typedef __attribute__((ext_vector_type(2))) float v2f;
typedef __attribute__((ext_vector_type(8))) float v8f;

#define BB 32
#define LL 12
#define NN 4096
#define EROWS (BB * LL)      // 384 GEMM rows
#define ETOT (EROWS * NN)    // 1572864 elements
#define HID 64
#define NSTEP 10
#define DT_ 0.1f

// ------------------------------------------------------------------
// 1) base_pred[b,l,n] = prior + tanh(Wad@(Wk@hist_c0)+bad)*kasa_gate
// ------------------------------------------------------------------
__global__ __launch_bounds__(256) void k_basepred(
    const float* __restrict__ hist, const float* __restrict__ Wk,
    const float* __restrict__ bk, const float* __restrict__ Wad,
    const float* __restrict__ bad, const float* __restrict__ kg,
    float* __restrict__ base)
{
  __shared__ float sWk[144], sWad[144], sbk[12], sbad[12];
  int tid = threadIdx.x;
  if (tid < 144) { sWk[tid] = Wk[tid]; sWad[tid] = Wad[tid]; }
  if (tid < 12)  { sbk[tid] = bk[tid]; sbad[tid] = bad[tid]; }
  __syncthreads();
  int gid = blockIdx.x * 256 + tid;   // over B*N
  int b = gid >> 12;
  int n = gid & 4095;
  float h0[12];
#pragma unroll
  for (int t = 0; t < 12; t++)
    h0[t] = hist[((b * 12 + t) * 4096 + n) * 4 + 0];
  float prior = hist[((b * 12 + 11) * 4096 + n) * 4 + 3];
  float gate = kg[0];
  float kas[12];
#pragma unroll
  for (int l = 0; l < 12; l++) {
    float acc = sbk[l];
#pragma unroll
    for (int t = 0; t < 12; t++) acc = __builtin_fmaf(sWk[l * 12 + t], h0[t], acc);
    kas[l] = acc;
  }
#pragma unroll
  for (int o = 0; o < 12; o++) {
    float acc = sbad[o];
#pragma unroll
    for (int l = 0; l < 12; l++) acc = __builtin_fmaf(sWad[o * 12 + l], kas[l], acc);
    base[(b * 12 + o) * 4096 + n] = __builtin_fmaf(tanhf(acc), gate, prior);
  }
}

// ------------------------------------------------------------------
// 2/5) C[384,4096] = alpha * A[384,4096] x (transB ? U^T : U)  (+addend)
// 128 threads = 4 waves; block tile 64x64; wave tile 16x64 (4 v8f accs).
// fp32 WMMA 16x16x4. K staged in 64-chunks; BOTH tiles stored
// K-contiguous per lane (stride 68) so every fragment is one aligned
// ds_load_b64. float4 global loads, software-pipelined ahead of compute.
// ------------------------------------------------------------------
__global__ __launch_bounds__(128) void k_gemm(
    const float* __restrict__ A, const float* __restrict__ Bm,
    const float* __restrict__ addend, const float* __restrict__ gate,
    float alphaC, float* __restrict__ C, int transB)
{
  __shared__ float As[64][68];    // As[m][k]
  __shared__ float BsT[64][68];   // BsT[n][k] = B[k][n]
  int tid = threadIdx.x, wave = tid >> 5, lane = tid & 31;
  int m0 = blockIdx.y * 64, n0 = blockIdx.x * 64;
  v8f acc[4];
#pragma unroll
  for (int nt = 0; nt < 4; nt++)
#pragma unroll
    for (int r = 0; r < 8; r++) acc[nt][r] = 0.0f;

  int l16 = lane & 15;           // M for A-frag, N for B-frag
  int kp = (lane >> 4) << 1;     // upper half-wave holds K+2, K+3

  // ---- prologue: stage chunk 0 ----
#pragma unroll
  for (int j = 0; j < 8; j++) {
    int idx = j * 128 + tid;
    int r = idx >> 4, c4 = (idx & 15) << 2;
    *(float4*)&As[r][c4] = *(const float4*)&A[(m0 + r) * 4096 + c4];
  }
  if (transB) {
#pragma unroll
    for (int j = 0; j < 8; j++) {
      int idx = j * 128 + tid;
      int nn = idx >> 4, k4 = (idx & 15) << 2;
      *(float4*)&BsT[nn][k4] = *(const float4*)&Bm[(n0 + nn) * 4096 + k4];
    }
  } else {
#pragma unroll
    for (int j = 0; j < 8; j++) {
      int idx = j * 128 + tid;
      int r = idx >> 4, c4 = (idx & 15) << 2;
      float4 t = *(const float4*)&Bm[r * 4096 + n0 + c4];
      BsT[c4 + 0][r] = t.x; BsT[c4 + 1][r] = t.y;
      BsT[c4 + 2][r] = t.z; BsT[c4 + 3][r] = t.w;
    }
  }

  for (int k0 = 0; k0 < 4096; k0 += 64) {
    __syncthreads();
    // ---- prefetch next chunk into registers (overlaps with WMMA) ----
    float4 regA[8], regB[8];
    bool more = (k0 + 64) < 4096;
    if (more) {
      int kn = k0 + 64;
#pragma unroll
      for (int j = 0; j < 8; j++) {
        int idx = j * 128 + tid;
        int r = idx >> 4, c4 = (idx & 15) << 2;
        regA[j] = *(const float4*)&A[(m0 + r) * 4096 + kn + c4];
      }
      if (transB) {
#pragma unroll
        for (int j = 0; j < 8; j++) {
          int idx = j * 128 + tid;
          int nn = idx >> 4, k4 = (idx & 15) << 2;
          regB[j] = *(const float4*)&Bm[(n0 + nn) * 4096 + kn + k4];
        }
      } else {
#pragma unroll
        for (int j = 0; j < 8; j++) {
          int idx = j * 128 + tid;
          int r = idx >> 4, c4 = (idx & 15) << 2;
          regB[j] = *(const float4*)&Bm[(kn + r) * 4096 + n0 + c4];
        }
      }
    }
    // ---- compute: 16 K-steps x 4 N-tiles of WMMA ----
#pragma unroll
    for (int kk = 0; kk < 64; kk += 4) {
      v2f a = *(const v2f*)&As[wave * 16 + l16][kk + kp];
#pragma unroll
      for (int nt = 0; nt < 4; nt++) {
        v2f bfr = *(const v2f*)&BsT[nt * 16 + l16][kk + kp];
        acc[nt] = __builtin_amdgcn_wmma_f32_16x16x4_f32(
            false, a, false, bfr, (short)0, acc[nt], false, false);
      }
    }
    __syncthreads();
    // ---- commit prefetched tile to LDS ----
    if (more) {
#pragma unroll
      for (int j = 0; j < 8; j++) {
        int idx = j * 128 + tid;
        int r = idx >> 4, c4 = (idx & 15) << 2;
        *(float4*)&As[r][c4] = regA[j];
      }
      if (transB) {
#pragma unroll
        for (int j = 0; j < 8; j++) {
          int idx = j * 128 + tid;
          int nn = idx >> 4, k4 = (idx & 15) << 2;
          *(float4*)&BsT[nn][k4] = regB[j];
        }
      } else {
#pragma unroll
        for (int j = 0; j < 8; j++) {
          int idx = j * 128 + tid;
          int r = idx >> 4, c4 = (idx & 15) << 2;
          BsT[c4 + 0][r] = regB[j].x; BsT[c4 + 1][r] = regB[j].y;
          BsT[c4 + 2][r] = regB[j].z; BsT[c4 + 3][r] = regB[j].w;
        }
      }
    }
  }

  float alpha = alphaC * (gate ? gate[0] : 1.0f);
  int mShift = (lane >> 4) * 8;   // C/D layout: row = r + 8*(lane>=16)
#pragma unroll
  for (int nt = 0; nt < 4; nt++)
#pragma unroll
    for (int r = 0; r < 8; r++) {
      int row = m0 + wave * 16 + r + mShift;
      int col = n0 + nt * 16 + l16;
      float v = alpha * acc[nt][r];
      if (addend) v += addend[row * 4096 + col];
      C[row * 4096 + col] = v;
    }
}

// ------------------------------------------------------------------
// 3) per-step constant t-feature hidden contribution (b1 folded in)
// ------------------------------------------------------------------
__global__ __launch_bounds__(256) void k_ttable(
    const float* __restrict__ A1, const float* __restrict__ B1,
    const float* __restrict__ b1, float* __restrict__ ttab)
{
  int idx = blockIdx.x * 256 + threadIdx.x;
  if (idx >= NSTEP * HID) return;
  int k = idx >> 6, h = idx & 63;
  float t = DT_ * (float)k;
  float acc = b1[h];
#pragma unroll
  for (int g = 0; g < 8; g++) {
    float ang = t * (float)(g + 1);
    acc += A1[h * 24 + 16 + g] * __cosf(ang) + B1[h * 24 + 16 + g] * __sinf(ang);
  }
  ttab[idx] = acc;
}

// ------------------------------------------------------------------
// 4) fused 10-step Euler ODE, one element per lane, all in registers.
// ------------------------------------------------------------------
__global__ __launch_bounds__(256) void k_ode(
    const float* __restrict__ x0, const float* __restrict__ cond,
    const float* __restrict__ A1, const float* __restrict__ B1,
    const float* __restrict__ W2, const float* __restrict__ b2,
    const float* __restrict__ ttab, float* __restrict__ xf)
{
  __shared__ float A1x[64][8], B1x[64][8], A1c[64][8], B1c[64][8];
  __shared__ float W2s[64], tts[10][64];
  int tid = threadIdx.x;
  for (int i = tid; i < 512; i += 256) {
    int h = i >> 3, g = i & 7;
    A1x[h][g] = A1[h * 24 + g];
    B1x[h][g] = B1[h * 24 + g];
    A1c[h][g] = A1[h * 24 + 8 + g];
    B1c[h][g] = B1[h * 24 + 8 + g];
  }
  if (tid < 64) W2s[tid] = W2[tid];
  for (int i = tid; i < NSTEP * 64; i += 256) tts[i >> 6][i & 63] = ttab[i];
  __syncthreads();

  int e = blockIdx.x * 256 + tid;
  float x = x0[e];
  float cs = cond[e];
  float b2v = b2[0];

  // cond-feature hidden contribution (reused by all 10 steps)
  float hc[64];
#pragma unroll
  for (int h = 0; h < 64; h++) hc[h] = 0.0f;
  {
    float cb = __cosf(cs), sb = __sinf(cs);
    float cg = cb, sg = sb;
    for (int g = 0; g < 8; g++) {
#pragma unroll
      for (int h = 0; h < 64; h++)
        hc[h] = __builtin_fmaf(A1c[h][g], cg, __builtin_fmaf(B1c[h][g], sg, hc[h]));
      float cn = cg * cb - sg * sb;
      float sn = sg * cb + cg * sb;
      cg = cn; sg = sn;
    }
  }

  for (int k = 0; k < NSTEP; k++) {
    float hl[64];
#pragma unroll
    for (int h = 0; h < 64; h++) hl[h] = hc[h] + tts[k][h];
    float cb = __cosf(x), sb = __sinf(x);
    float cg = cb, sg = sb;
    for (int g = 0; g < 8; g++) {
#pragma unroll
      for (int h = 0; h < 64; h++)
        hl[h] = __builtin_fmaf(A1x[h][g], cg, __builtin_fmaf(B1x[h][g], sg, hl[h]));
      float cn = cg * cb - sg * sb;
      float sn = sg * cb + cg * sb;
      cg = cn; sg = sn;
    }
    float v = b2v;
#pragma unroll
    for (int h = 0; h < 64; h++) {
      float z = hl[h];
      float s = z * __builtin_amdgcn_rcpf(1.0f + __expf(-z));  // silu
      v = __builtin_fmaf(s, W2s[h], v);
    }
    x = __builtin_fmaf(DT_, v, x);
  }
  xf[e] = x;   // sqrt(N)=64 scaling folded into final GEMM alpha
}

// ------------------------------------------------------------------
extern "C" void kernel_launch(void* const* d_in, const int* in_sizes, int n_in,
                              void* d_out, int out_size, void* d_ws, size_t ws_size,
                              hipStream_t stream)
{
  const float* hist = (const float*)d_in[0];
  const float* x0v  = (const float*)d_in[1];
  const float* U    = (const float*)d_in[2];
  const float* Wk   = (const float*)d_in[3];
  const float* bk   = (const float*)d_in[4];
  const float* Wad  = (const float*)d_in[5];
  const float* bad  = (const float*)d_in[6];
  const float* kg   = (const float*)d_in[7];
  const float* fg   = (const float*)d_in[8];
  const float* A1   = (const float*)d_in[9];
  const float* B1   = (const float*)d_in[10];
  const float* b1   = (const float*)d_in[11];
  const float* W2   = (const float*)d_in[12];
  const float* b2   = (const float*)d_in[13];
  float* out = (float*)d_out;
  float* ws  = (float*)d_ws;

  float* base = ws;                         // [384,4096]
  float* cond = ws + (size_t)ETOT;          // [384,4096]
  float* xfin = ws + (size_t)2 * ETOT;      // [384,4096]
  float* ttab = ws + (size_t)3 * ETOT;      // [10,64]

  // 1) base_pred
  k_basepred<<<(BB * NN) / 256, 256, 0, stream>>>(hist, Wk, bk, Wad, bad, kg, base);
  // 2) cond = (base @ U) / 64
  k_gemm<<<dim3(NN / 64, EROWS / 64), 128, 0, stream>>>(
      base, U, nullptr, nullptr, 1.0f / 64.0f, cond, 0);
  // 3) t-feature table
  k_ttable<<<(NSTEP * HID + 255) / 256, 256, 0, stream>>>(A1, B1, b1, ttab);
  // 4) fused Euler ODE
  k_ode<<<ETOT / 256, 256, 0, stream>>>(x0v, cond, A1, B1, W2, b2, ttab, xfin);
  // 5) out = base + fm_gate * 64 * (xfin @ U^T)
  k_gemm<<<dim3(NN / 64, EROWS / 64), 128, 0, stream>>>(
      xfin, U, base, fg, 64.0f, out, 1);
}